// ProposalLayer_26508538151745
// MI455X (gfx1250) — compile-verified
//
#include <hip/hip_runtime.h>

typedef __attribute__((ext_vector_type(2))) float v2f;
typedef __attribute__((ext_vector_type(8))) float v8f;
typedef __attribute__((ext_vector_type(4))) unsigned int v4u;
typedef __attribute__((ext_vector_type(8))) int v8i;
typedef __attribute__((ext_vector_type(4))) int v4i;

#define B_DIM 256
#define P_DIM 1024
#define G_DIM 128
#define CCENTER 4000.0f           // COORD_SCALE/2, reduces fp32 cancellation
#define THRESH2 (500.0f * 500.0f) // DIST_THRESH^2
#define BBOX_TH 0.1f

// One wave (32 lanes) handles a 16-proposal x 128-GT tile via 8x
// V_WMMA_F32_16X16X4_F32 (K=3 padded to 4).  Block = 8 waves = 128 proposals.
// gt_3d[b] (384 dwords) is staged into LDS once per block by the TDM
// (tensor_load_to_lds), then pre-swizzled into WMMA B-fragment layout.
__global__ __launch_bounds__(256) void proposal_layer_kernel(
    const float* __restrict__ topk_index,   // [B,P,3]
    const float* __restrict__ topk_confs,   // [B,P]
    const float* __restrict__ bbox_preds,   // [B,P,2]
    const float* __restrict__ gt_3d,        // [B,G,3]
    const float* __restrict__ gt_bbox,      // [B,G,2]
    const int*   __restrict__ num_person,   // [B]
    float* __restrict__ out)                // [B,P,7]
{
    __shared__ __align__(16) float  s_raw[G_DIM * 3]; // TDM dest: gt_3d[b]
    __shared__ __align__(8)  float2 s_bf[2 * G_DIM];  // B-frags: [half*128+col]
    __shared__               float  s_gsq[G_DIM];     // ||g-c||^2 per GT

    const int b     = blockIdx.y;
    const int wave  = threadIdx.x >> 5;          // 0..7
    const int lane  = threadIdx.x & 31;
    const int l16   = lane & 15;
    const bool lo   = lane < 16;
    const int pbase = (blockIdx.x * 8 + wave) * 16;
    const int np    = num_person[b];

    // ---- A fragment (global load issued early, overlaps the TDM DMA) ----
    // 16x4 f32: lanes 0-15 hold (K0,K1) of row M=l16; lanes 16-31 hold (K2,K3).
    // A[M][K] = -2*(x_M[K]-c), K=3 padded with 0.
    v2f afrag;
    {
        const float* xp = topk_index + ((size_t)(b * P_DIM + pbase + l16)) * 3;
        float ax = -2.0f * (xp[0] - CCENTER);
        float ay = -2.0f * (xp[1] - CCENTER);
        float az = -2.0f * (xp[2] - CCENTER);
        afrag.x = lo ? ax : az;
        afrag.y = lo ? ay : 0.0f;
    }

    // ---- TDM: DMA gt_3d[b] (1 row x 384 dwords) into LDS (wave 0 only) ----
    if (wave == 0) {
        const unsigned long long ga =
            (unsigned long long)(uintptr_t)(gt_3d + (size_t)b * G_DIM * 3);
        const unsigned int la = (unsigned int)(uintptr_t)(&s_raw[0]);
        // D# group0: count=1 | lds_addr | global_addr[56:0] | type=2
        v4u g0;
        g0.x = 1u;
        g0.y = la;
        g0.z = (unsigned int)ga;
        g0.w = ((unsigned int)(ga >> 32) & 0x01FFFFFFu) | 0x80000000u;
        // D# group1: data_size=4B; tensor_dim0=384, tensor_dim1=1;
        //            tile_dim0=384, tile_dim1=1; tensor_dim0_stride=384
        v8i g1;
        g1.s0 = 0x00020000;            // data_size=2 (4 bytes) at [17:16]
        g1.s1 = (int)(384u << 16);     // tensor_dim0[15:0] at [63:48]
        g1.s2 = (int)(1u << 16);       // tensor_dim1[15:0] at [95:80]
        g1.s3 = (int)(384u << 16);     // tile_dim0 at [127:112]
        g1.s4 = 1;                     // tile_dim1=1 at [143:128]
        g1.s5 = 384;                   // tensor_dim0_stride[31:0] at [191:160]
        g1.s6 = 0;
        g1.s7 = 0;
        v4i gz4 = {0, 0, 0, 0};        // groups 2/3 unused (<=2D tensor)
        v8i gz8 = {0, 0, 0, 0, 0, 0, 0, 0};
        __builtin_amdgcn_tensor_load_to_lds(g0, g1, gz4, gz4, gz8, 0);
        __builtin_amdgcn_s_wait_tensorcnt(0);
    }
    __syncthreads();

    // ---- pre-swizzle B fragments + ||g||^2 (cooperative, 256 threads) ----
    {
        const int col = threadIdx.x & 127;
        const float cx = s_raw[col * 3 + 0] - CCENTER;
        const float cy = s_raw[col * 3 + 1] - CCENTER;
        const float cz = s_raw[col * 3 + 2] - CCENTER;
        if (threadIdx.x < 128) {
            s_bf[col]  = make_float2(cx, cy);         // rows K0,K1 (lanes 0-15)
            s_gsq[col] = cx * cx + cy * cy + cz * cz;
        } else {
            s_bf[G_DIM + col] = make_float2(cz, 0.0f); // rows K2,K3 (lanes 16-31)
        }
    }
    __syncthreads();

    // ---- main loop: 8 WMMA tiles over the 128 GT columns ----
    float bs[8];
    int   bi[8];
#pragma unroll
    for (int v = 0; v < 8; ++v) { bs[v] = __builtin_inff(); bi[v] = 0; }

    const v8f czero = {};
    const int bofs  = lo ? 0 : G_DIM;

#pragma unroll
    for (int t = 0; t < 8; ++t) {
        const int col = t * 16 + l16;            // GT column this lane owns
        const float2 bf2 = s_bf[bofs + col];     // one ds_load_b64
        v2f bfrag; bfrag.x = bf2.x; bfrag.y = bf2.y;
        const float gsq = s_gsq[col];            // one ds_load_b32

        // D[M][N] = -2 * (x_M . g_N)   (EXEC all 1s here)
        v8f d = __builtin_amdgcn_wmma_f32_16x16x4_f32(
            false, afrag, false, bfrag, (short)0, czero, false, false);

        const bool valid = col < np;             // mask padded GT slots
#pragma unroll
        for (int v = 0; v < 8; ++v) {
            float s = valid ? (gsq + d[v]) : __builtin_inff();
            if (s < bs[v]) { bs[v] = s; bi[v] = col; }  // strict <: first-min wins
        }
    }

    // ---- cross-lane argmin within 16-lane halves (rows stay in-half) ----
#pragma unroll
    for (int v = 0; v < 8; ++v) {
        float s = bs[v]; int i = bi[v];
#pragma unroll
        for (int m = 8; m >= 1; m >>= 1) {
            float s2 = __shfl_xor(s, m, 32);
            int   i2 = __shfl_xor(i, m, 32);
            if (s2 < s || (s2 == s && i2 < i)) { s = s2; i = i2; }
        }
        bs[v] = s; bi[v] = i;
    }

    // Lane l16==v owns row slot v (rows 0-7 in lower half, 8-15 in upper).
    float fsc = 0.0f; int fidx = 0;
#pragma unroll
    for (int v = 0; v < 8; ++v) {
        if (l16 == v) { fsc = bs[v]; fidx = bi[v]; }
    }

    if (l16 < 8) {
        const int row = pbase + (lo ? l16 : l16 + 8);   // proposal index
        const float* xp = topk_index + ((size_t)(b * P_DIM + row)) * 3;
        const float x = xp[0], y = xp[1], z = xp[2];
        const float xc = x - CCENTER, yc = y - CCENTER, zc = z - CCENTER;
        const float d2 = xc * xc + yc * yc + zc * zc + fsc;   // min dist^2
        const float p2g = (d2 > THRESH2) ? -1.0f : (float)fidx;

        const float* gb = gt_bbox + ((size_t)(b * G_DIM + fidx)) * 2;
        const float mb0 = gb[0], mb1 = gb[1];
        const float* bp = bbox_preds + ((size_t)(b * P_DIM + row)) * 2;
        const float p0 = bp[0], p1 = bp[1];
        const bool cond = (p2g >= 0.0f) &&
                          ((p0 < mb0 - BBOX_TH) || (p1 < mb1 - BBOX_TH));
        const float o5 = cond ? mb0 : p0;
        const float o6 = cond ? mb1 : p1;
        const float conf = topk_confs[(size_t)b * P_DIM + row];

        float* op = out + ((size_t)(b * P_DIM + row)) * 7;
        op[0] = x;   op[1] = y;   op[2] = z;
        op[3] = p2g; op[4] = conf;
        op[5] = o5;  op[6] = o6;
    }
}

extern "C" void kernel_launch(void* const* d_in, const int* in_sizes, int n_in,
                              void* d_out, int out_size, void* d_ws, size_t ws_size,
                              hipStream_t stream) {
    (void)in_sizes; (void)n_in; (void)d_ws; (void)ws_size; (void)out_size;
    const float* topk_index  = (const float*)d_in[0];
    const float* topk_confs  = (const float*)d_in[1];
    const float* bbox_preds  = (const float*)d_in[2];
    const float* gt_3d       = (const float*)d_in[3];
    const float* gt_bbox     = (const float*)d_in[4];
    const int*   num_person  = (const int*)d_in[5];
    float* out = (float*)d_out;

    dim3 grid(P_DIM / 128, B_DIM);   // 8 proposal-chunks x 256 batches
    dim3 block(256);                 // 8 waves, each = 16x128 WMMA tile
    proposal_layer_kernel<<<grid, block, 0, stream>>>(
        topk_index, topk_confs, bbox_preds, gt_3d, gt_bbox, num_person, out);
}